// GALAProjection_44796508897351
// MI455X (gfx1250) — compile-verified
//
#include <hip/hip_runtime.h>
#include <hip/hip_bf16.h>
#include <math.h>

// ---------------------------------------------------------------------------
// GALAProjection for MI455X (gfx1250, wave32).
//   Stage 1: pf = point_features @ W_feat.T + b_feat   -> WMMA f32 16x16x4
//   Stage 2: brute-force KNN (top-32 of 20000 per grid point), LDS-staged
//   Stage 3: per-neighbor MLP + softmax + weighted gather-aggregate (L2-bound)
//   Stage 4: out = agg @ W_out.T + b_out, fused LayerNorm  -> WMMA f32 16x16x4
// ---------------------------------------------------------------------------

typedef __attribute__((ext_vector_type(2))) float v2f;
typedef __attribute__((ext_vector_type(8))) float v8f;

#define HH 96
#define WW 96
#define DHW (HH * WW)        // 9216
#define NPTS 20000
#define KNN 32
#define PD 64
#define GD 256
#define HID 64
#define LN_EPS 1e-5f

// ---------------------------------------------------------------------------
// Kernel 1: pf[20000,256] = PF[20000,64] @ W_feat[256,64]^T + b_feat
// One wave per 16x16 output tile; K=64 swept in 16 WMMA steps of 4.
// A layout (16x4 f32): lanes 0-15 -> {K0,K1}, lanes 16-31 -> {K2,K3}, M=lane&15
// B layout (4x16 f32): mirrored; column = lane&15; contiguous 2-float loads
//                      from W_feat rows since B[k][n] = W_feat[n][k].
// ---------------------------------------------------------------------------
__global__ __launch_bounds__(32)
void k_pf_gemm(const float* __restrict__ PF, const float* __restrict__ Wf,
               const float* __restrict__ bf, float* __restrict__ pf) {
  const int tile_m = blockIdx.x;      // 0..1249 (20000/16)
  const int tile_n = blockIdx.y;      // 0..15   (256/16)
  const int lane = threadIdx.x;
  const int hi = lane >> 4, lo = lane & 15;
  const int m0 = tile_m * 16, n0 = tile_n * 16;

  const float* arow = PF + (size_t)(m0 + lo) * PD + 2 * hi;
  const float* brow = Wf + (size_t)(n0 + lo) * PD + 2 * hi;

  v8f acc = {};
#pragma unroll
  for (int kk = 0; kk < PD; kk += 4) {
    v2f av = { arow[kk], arow[kk + 1] };
    v2f bv = { brow[kk], brow[kk + 1] };
    acc = __builtin_amdgcn_wmma_f32_16x16x4_f32(false, av, false, bv,
                                                (short)0, acc, false, false);
  }

  const float bias = bf[n0 + lo];
#pragma unroll
  for (int r = 0; r < 8; ++r) {
    pf[(size_t)(m0 + r + 8 * hi) * GD + (n0 + lo)] = acc[r] + bias;
  }
}

// ---------------------------------------------------------------------------
// Kernel 2: exact KNN. 128 threads/block, one grid point per thread.
// Point coords staged through LDS in 2048-point chunks (16 KB); each thread
// keeps its 32-candidate list in LDS with register-tracked current max.
// ---------------------------------------------------------------------------
#define KNN_BLK 128
#define KNN_CHUNK 2048

__global__ __launch_bounds__(KNN_BLK)
void k_knn(const float* __restrict__ pc, int* __restrict__ idx_out) {
  __shared__ float sc[KNN_CHUNK * 2];          // 16 KB coords chunk
  __shared__ float bdist[KNN_BLK * KNN];       // 16 KB candidate dists
  __shared__ int   bidx [KNN_BLK * KNN];       // 16 KB candidate indices

  const int tid = threadIdx.x;
  const int g = blockIdx.x * KNN_BLK + tid;    // 0..9215
  const int iy = g / WW, ix = g % WW;
  const float gx = (float)ix * (1.0f / (WW - 1));
  const float gy = (float)iy * (1.0f / (HH - 1));

  const int lbase = tid * KNN;
  for (int s = 0; s < KNN; ++s) bdist[lbase + s] = 1e30f;
  float curmax = 1e30f;
  int carg = 0;

  for (int base = 0; base < NPTS; base += KNN_CHUNK) {
    const int cn = min(KNN_CHUNK, NPTS - base);
    __syncthreads();
    for (int i = tid; i < cn * 2; i += KNN_BLK) sc[i] = pc[base * 2 + i];
    __syncthreads();

    for (int j = 0; j < cn; ++j) {
      const float dx = gx - sc[2 * j];
      const float dy = gy - sc[2 * j + 1];
      const float d = dx * dx + dy * dy;
      if (d < curmax) {
        bdist[lbase + carg] = d;
        bidx [lbase + carg] = base + j;
        float m = -1.0f; int a = 0;
#pragma unroll
        for (int s = 0; s < KNN; ++s) {
          const float v = bdist[lbase + s];
          if (v > m) { m = v; a = s; }
        }
        curmax = m; carg = a;
      }
    }
  }

  for (int s = 0; s < KNN; ++s) idx_out[(size_t)g * KNN + s] = bidx[lbase + s];
}

// ---------------------------------------------------------------------------
// Kernel 3: per-grid-point attention MLP + softmax + weighted feature gather.
// One block (256 threads) per grid point; pf gathers hit L2 (20.5 MB table).
// ---------------------------------------------------------------------------
__global__ __launch_bounds__(256)
void k_attn_agg(const float* __restrict__ pc,
                const float* __restrict__ W1, const float* __restrict__ b1,
                const float* __restrict__ W2, const float* __restrict__ b2,
                const int* __restrict__ idx, const float* __restrict__ pf,
                float* __restrict__ agg) {
  __shared__ float sW1[HID * 3];
  __shared__ float sb1[HID];
  __shared__ float sW2[HID];
  __shared__ float slog[KNN];
  __shared__ float sw[KNN];
  __shared__ int   sidx[KNN];
  __shared__ float smax, sden;

  const int g = blockIdx.x;
  const int tid = threadIdx.x;
  const int iy = g / WW, ix = g % WW;
  const float gx = (float)ix * (1.0f / (WW - 1));
  const float gy = (float)iy * (1.0f / (HH - 1));

  if (tid < HID * 3) sW1[tid] = W1[tid];
  else if (tid < HID * 3 + HID) {
    const int h = tid - HID * 3;
    sb1[h] = b1[h];
    sW2[h] = W2[h];
  }
  if (tid < KNN) sidx[tid] = idx[(size_t)g * KNN + tid];
  __syncthreads();

  if (tid < KNN) {
    const int p = sidx[tid];
    const float rx = gx - pc[2 * p];
    const float ry = gy - pc[2 * p + 1];
    const float dn = sqrtf(rx * rx + ry * ry);
    float acc = b2[0];
#pragma unroll 8
    for (int h = 0; h < HID; ++h) {
      const float z = rx * sW1[h * 3 + 0] + ry * sW1[h * 3 + 1] +
                      dn * sW1[h * 3 + 2] + sb1[h];
      const float ge = 0.5f * z * (1.0f + erff(z * 0.70710678118654752f));
      acc += sW2[h] * ge;
    }
    slog[tid] = acc;
  }
  __syncthreads();

  if (tid == 0) {
    float m = slog[0];
    for (int k = 1; k < KNN; ++k) m = fmaxf(m, slog[k]);
    float den = 0.0f;
    for (int k = 0; k < KNN; ++k) den += expf(slog[k] - m);
    smax = m; sden = den;
  }
  __syncthreads();
  if (tid < KNN) sw[tid] = expf(slog[tid] - smax) / sden;
  __syncthreads();

  // Weighted aggregation: thread = output channel, coalesced gather per k.
  float a = 0.0f;
#pragma unroll 4
  for (int k = 0; k < KNN; ++k) {
    const float* row = pf + (size_t)sidx[k] * GD;
    if (k + 1 < KNN)
      __builtin_prefetch(pf + (size_t)sidx[k + 1] * GD + tid, 0, 1);
    a += sw[k] * row[tid];
  }
  agg[(size_t)g * GD + tid] = a;
}

// ---------------------------------------------------------------------------
// Kernel 4: out = agg @ W_out^T + b_out, fused LayerNorm over GD=256.
// 512 threads = 16 waves; wave w computes cols w*16..w*16+15 of a 16-row
// tile via 64 WMMA f32 16x16x4 steps; full 16x256 tile lands in LDS for LN.
// ---------------------------------------------------------------------------
__global__ __launch_bounds__(512)
void k_out_ln(const float* __restrict__ agg, const float* __restrict__ Wo,
              const float* __restrict__ bo, const float* __restrict__ lng,
              const float* __restrict__ lnb, float* __restrict__ out) {
  __shared__ float tile[16][GD + 4];   // +4 pad vs bank conflicts
  __shared__ float smu[16], srs[16];

  const int m0 = blockIdx.x * 16;      // 9216/16 = 576 blocks
  const int tid = threadIdx.x;
  const int lane = tid & 31, wv = tid >> 5;
  const int hi = lane >> 4, lo = lane & 15;
  const int n0 = wv * 16;

  const float* arow = agg + (size_t)(m0 + lo) * GD + 2 * hi;
  const float* brow = Wo + (size_t)(n0 + lo) * GD + 2 * hi;  // B[k][n]=Wo[n][k]

  v8f acc = {};
#pragma unroll
  for (int kk = 0; kk < GD; kk += 4) {
    v2f av = { arow[kk], arow[kk + 1] };
    v2f bv = { brow[kk], brow[kk + 1] };
    acc = __builtin_amdgcn_wmma_f32_16x16x4_f32(false, av, false, bv,
                                                (short)0, acc, false, false);
  }

  const float bias = bo[n0 + lo];
#pragma unroll
  for (int r = 0; r < 8; ++r) tile[r + 8 * hi][n0 + lo] = acc[r] + bias;
  __syncthreads();

  if (tid < 16) {
    float s = 0.0f;
    for (int c = 0; c < GD; ++c) s += tile[tid][c];
    const float mu = s * (1.0f / GD);
    float v = 0.0f;
    for (int c = 0; c < GD; ++c) {
      const float d = tile[tid][c] - mu;
      v += d * d;
    }
    smu[tid] = mu;
    srs[tid] = rsqrtf(v * (1.0f / GD) + LN_EPS);
  }
  __syncthreads();

  for (int e = tid; e < 16 * GD; e += 512) {
    const int r = e >> 8, c = e & (GD - 1);
    out[(size_t)(m0 + r) * GD + c] =
        (tile[r][c] - smu[r]) * srs[r] * lng[c] + lnb[c];
  }
}

// ---------------------------------------------------------------------------
// Host launch
// ---------------------------------------------------------------------------
extern "C" void kernel_launch(void* const* d_in, const int* in_sizes, int n_in,
                              void* d_out, int out_size, void* d_ws, size_t ws_size,
                              hipStream_t stream) {
  const float* point_features = (const float*)d_in[0];   // [20000,64]
  const float* point_coords   = (const float*)d_in[1];   // [20000,2]
  const float* W_feat         = (const float*)d_in[2];   // [256,64]
  const float* b_feat         = (const float*)d_in[3];   // [256]
  const float* W1             = (const float*)d_in[4];   // [64,3]
  const float* b1             = (const float*)d_in[5];   // [64]
  const float* W2             = (const float*)d_in[6];   // [1,64]
  const float* b2             = (const float*)d_in[7];   // [1]
  const float* W_out          = (const float*)d_in[8];   // [256,256]
  const float* b_out          = (const float*)d_in[9];   // [256]
  const float* ln_g           = (const float*)d_in[10];  // [256]
  const float* ln_b           = (const float*)d_in[11];  // [256]
  float* out = (float*)d_out;                            // [9216,256]

  // Workspace layout (floats): pf | idx | agg  (~31 MB total)
  float* ws = (float*)d_ws;
  float* pf  = ws;                                        // 20000*256
  int*   idx = (int*)(ws + (size_t)NPTS * GD);            // 9216*32
  float* agg = ws + (size_t)NPTS * GD + (size_t)DHW * KNN;// 9216*256

  k_pf_gemm<<<dim3(NPTS / 16, GD / 16), 32, 0, stream>>>(point_features, W_feat,
                                                         b_feat, pf);
  k_knn<<<DHW / KNN_BLK, KNN_BLK, 0, stream>>>(point_coords, idx);
  k_attn_agg<<<DHW, 256, 0, stream>>>(point_coords, W1, b1, W2, b2, idx, pf, agg);
  k_out_ln<<<DHW / 16, 512, 0, stream>>>(agg, W_out, b_out, ln_g, ln_b, out);
}